// SharpCosSim2D_78348793413637
// MI455X (gfx1250) — compile-verified
//
#include <hip/hip_runtime.h>
#include <math.h>

// Problem constants (match reference)
#define BN 16
#define HN 256
#define WN 256
#define CN 64
#define UN 64
#define NROWS (BN * HN * WN)   // 1,048,576 rows of length CN
#define EPSF 1e-12f

#define ROWS_PER_WAVE 16
#define WAVES_PER_BLOCK 8
#define ROWS_PER_BLOCK (ROWS_PER_WAVE * WAVES_PER_BLOCK)  // 128

typedef __attribute__((ext_vector_type(2))) float v2f;
typedef __attribute__((ext_vector_type(8))) float v8f;

// ---------------------------------------------------------------------------
// Prep: column-normalize w -> wn_t stored TRANSPOSED [u][c]; exponent & b^2.
// ---------------------------------------------------------------------------
__global__ void sharpcos_prep(const float* __restrict__ w,   // [CN][UN]
                              const float* __restrict__ b,   // [UN]
                              const float* __restrict__ p,   // [UN]
                              float* __restrict__ wn_t,      // [UN][CN]
                              float* __restrict__ expo,      // [UN]
                              float* __restrict__ bsq) {     // [UN]
    int u = threadIdx.x;
    if (u >= UN) return;
    float s = 0.f;
    for (int c = 0; c < CN; ++c) {
        float v = w[c * UN + u];
        s += v * v;
    }
    float inv = rsqrtf(fmaxf(s, EPSF));
    for (int c = 0; c < CN; ++c)
        wn_t[u * CN + c] = w[c * UN + u] * inv;

    float pv  = p[u];
    float sig = 1.f / (1.f + expf(-pv));
    float sp  = (pv > 20.f) ? pv : log1pf(expf(pv));   // softplus
    expo[u]   = sig * sp;
    float bv  = b[u];
    bsq[u]    = bv * bv;
}

// ---------------------------------------------------------------------------
// Main: each wave computes a 16x64 output tile via V_WMMA_F32_16X16X4_F32.
// Row L2-normalization is folded into the epilogue (matmul is linear in x).
// ---------------------------------------------------------------------------
__global__ __launch_bounds__(256)
void sharpcos_main(const float* __restrict__ x,      // [NROWS][CN]
                   const float* __restrict__ wn_t,   // [UN][CN]
                   const float* __restrict__ expo,   // [UN]
                   const float* __restrict__ bsq,    // [UN]
                   float* __restrict__ out) {        // [NROWS][UN]
    __shared__ float lds_wnt[UN * CN];            // 16 KB
    __shared__ float lds_x[ROWS_PER_BLOCK * CN];  // 32 KB
    __shared__ float lds_rinv[ROWS_PER_BLOCK];

    const int tid  = threadIdx.x;
    const int lane = tid & 31;
    const int wave = tid >> 5;
    const int hf   = lane >> 4;    // 0: lanes 0-15, 1: lanes 16-31
    const int l15  = lane & 15;
    const int block_row0 = blockIdx.x * ROWS_PER_BLOCK;

    // Stage normalized-W (transposed) into LDS: 1024 float4, coalesced.
    {
        const float4* src = (const float4*)wn_t;
        float4*       dst = (float4*)lds_wnt;
        #pragma unroll
        for (int i = 0; i < (UN * CN) / 4 / 256; ++i)
            dst[tid + i * 256] = src[tid + i * 256];
    }
    // Stage 128x64 x tile into LDS: 2048 float4, coalesced.
    {
        const float4* src = (const float4*)(x + (size_t)block_row0 * CN);
        float4*       dst = (float4*)lds_x;
        #pragma unroll
        for (int i = 0; i < (ROWS_PER_BLOCK * CN) / 4 / 256; ++i)
            dst[tid + i * 256] = src[tid + i * 256];
    }
    __syncthreads();

    // Per-row inverse L2 norm (threads 0..127, one row each).
    if (tid < ROWS_PER_BLOCK) {
        const float4* row = (const float4*)(lds_x + tid * CN);
        float s = 0.f;
        #pragma unroll
        for (int i = 0; i < CN / 4; ++i) {
            float4 v = row[i];
            s += v.x * v.x + v.y * v.y + v.z * v.z + v.w * v.w;
        }
        lds_rinv[tid] = rsqrtf(fmaxf(s, EPSF));
    }

    // Per-lane epilogue params (column u = nt*16 + l15), overlap with barrier.
    float ex[4], bq[4];
    #pragma unroll
    for (int nt = 0; nt < 4; ++nt) {
        ex[nt] = expo[nt * 16 + l15];
        bq[nt] = bsq[nt * 16 + l15];
    }
    __syncthreads();

    // --- WMMA main loop: D(16x16) += A(16x4) x B(4x16), 16 K-steps, 4 N-tiles
    // A frag VGPR j, lane l: x[wrow0 + l15][4*ks + 2*hf + j]      (b64 LDS read)
    // B frag VGPR j, lane l: wn_t[nt*16 + l15][4*ks + 2*hf + j]   (b64 LDS read)
    const int wrow0 = wave * ROWS_PER_WAVE;
    const float* ax = lds_x + (wrow0 + l15) * CN + 2 * hf;
    const float* bw = lds_wnt + l15 * CN + 2 * hf;

    v8f acc0 = {}, acc1 = {}, acc2 = {}, acc3 = {};
    #pragma unroll
    for (int ks = 0; ks < 16; ++ks) {
        v2f a  = *(const v2f*)(ax + 4 * ks);
        v2f b0 = *(const v2f*)(bw + 0 * 16 * CN + 4 * ks);
        v2f b1 = *(const v2f*)(bw + 1 * 16 * CN + 4 * ks);
        v2f b2 = *(const v2f*)(bw + 2 * 16 * CN + 4 * ks);
        v2f b3 = *(const v2f*)(bw + 3 * 16 * CN + 4 * ks);
        acc0 = __builtin_amdgcn_wmma_f32_16x16x4_f32(false, a, false, b0, (short)0, acc0, false, false);
        acc1 = __builtin_amdgcn_wmma_f32_16x16x4_f32(false, a, false, b1, (short)0, acc1, false, false);
        acc2 = __builtin_amdgcn_wmma_f32_16x16x4_f32(false, a, false, b2, (short)0, acc2, false, false);
        acc3 = __builtin_amdgcn_wmma_f32_16x16x4_f32(false, a, false, b3, (short)0, acc3, false, false);
    }

    // --- Epilogue: scale by 1/||x||, sign * pow(|y| + eps + b^2, exponent).
    // C/D layout: element r of lane l is row M = r + 8*hf, col = nt*16 + l15.
    // pow(base, e) = exp2(e * log2(base)) via hardware TRANS ops.
    float* outbase = out + (size_t)(block_row0 + wrow0) * UN;
    #pragma unroll
    for (int r = 0; r < 8; ++r) {
        const int   m    = r + 8 * hf;
        const float rinv = lds_rinv[wrow0 + m];
        float ys[4] = { acc0[r], acc1[r], acc2[r], acc3[r] };
        #pragma unroll
        for (int nt = 0; nt < 4; ++nt) {
            float y    = ys[nt] * rinv;
            float s    = (y > 0.f) ? 1.f : ((y < 0.f) ? -1.f : 0.f);
            float base = fabsf(y) + EPSF + bq[nt];          // strictly > 0
            float v    = s * __builtin_amdgcn_exp2f(ex[nt] * __builtin_amdgcn_logf(base));
            outbase[(size_t)m * UN + nt * 16 + l15] = v;
        }
    }
}

// ---------------------------------------------------------------------------
extern "C" void kernel_launch(void* const* d_in, const int* in_sizes, int n_in,
                              void* d_out, int out_size, void* d_ws, size_t ws_size,
                              hipStream_t stream) {
    const float* x = (const float*)d_in[0];   // (16,256,256,64)
    const float* w = (const float*)d_in[1];   // (1,64,64)
    const float* b = (const float*)d_in[2];   // (64,)
    const float* p = (const float*)d_in[3];   // (64,)
    float* out = (float*)d_out;

    float* wn_t = (float*)d_ws;               // [UN][CN] = 4096 floats
    float* expo = wn_t + UN * CN;             // 64 floats
    float* bsq  = expo + UN;                  // 64 floats

    sharpcos_prep<<<1, 64, 0, stream>>>(w, b, p, wn_t, expo, bsq);
    sharpcos_main<<<NROWS / ROWS_PER_BLOCK, 256, 0, stream>>>(x, wn_t, expo, bsq, out);
}